// KANLayer_68487548502341
// MI455X (gfx1250) — compile-verified
//
#include <hip/hip_runtime.h>
#include <math.h>

// KAN layer for MI455X (gfx1250, wave32).
// Shapes: NUM_IN=NUM_OUT=128, SIZE=16384, BATCH=512, 8 basis fns (G+K), 6 knots.
// Bandwidth-bound: ~101MB of output writes -> design for coalesced stores.
// Spline eval factorized as per-input-feature GEMM Coef(16o x 12j) x Basis(12j x 16b)
// using V_WMMA_F32_16X16X4_F32 (K=12 -> 3 wmma steps; post_splines uses K=8 -> 2 steps).
// v3: Cox-de Boor divisions via v_rcp_f32 (one reciprocal per distinct knot-difference,
// shared between adjacent basis terms) -- __fdividef lowered to IEEE div on this
// toolchain and division was co-critical with store bandwidth.

#define NUM_IN   128
#define SIZE_S   16384
#define BATCH    512
#define NJ       8
#define NKNOT    6
#define PRE_ELEMS 8388608   // 512*16384

typedef float v2f __attribute__((ext_vector_type(2)));
typedef float v8f __attribute__((ext_vector_type(8)));

// LDS staging: element (il in [0,32), oo in [0,16), b in [0,16))
// addr = il*321 + oo*20 + b  (floats). 321 = 1 mod 64 -> store-pass reads
// (lanes vary il) are bank-conflict free; oo stride 20 -> the two D-tile
// halves (oo and oo+8) write disjoint bank ranges.
#define IL_STRIDE 321
#define OO_STRIDE 20
#define BUF_FLOATS (31*IL_STRIDE + 15*OO_STRIDE + 16 + 32) // ~10.3K floats, ~41KB

__global__ __launch_bounds__(256)
void kan_main(const float* __restrict__ x,
              const float* __restrict__ scale_b,
              const float* __restrict__ scale_spline,
              const float* __restrict__ coef,
              const float* __restrict__ mask,
              const float* __restrict__ knots,
              float* __restrict__ out_pre,
              float* __restrict__ out_spl,
              float* __restrict__ out_act,
              float* __restrict__ ws_part)
{
    __shared__ float buf[BUF_FLOATS];

    const int tid   = threadIdx.x;
    const int lane  = tid & 31;
    const int wave  = tid >> 5;
    const int m     = lane & 15;   // A-matrix row (o in tile) == B/D column (b in tile)
    const int half  = lane >> 4;   // K-pairing half per WMMA f32 register layout
    const int o_tile = blockIdx.x & 7;
    const int b_tile = (blockIdx.x >> 3) & 31;
    const int ihalf  = blockIdx.x >> 8;      // 0 or 1: which half of the i range
    const int o_base = o_tile * 16;
    const int b_base = b_tile * 16;

    v8f d2h[4];   // post_acts D tiles held across the staging phases

    for (int kk = 0; kk < 2; ++kk) {        // i rounds: i = k*32 + il, il in [0,32)
        const int k = ihalf * 2 + kk;
        // ---- compute: each wave produces D tiles for 4 consecutive il ----
        #pragma unroll
        for (int c = 0; c < 4; ++c) {
            const int il = wave * 4 + c;
            const int i  = k * 32 + il;

            // --- B-spline basis for (i, b=b_base+m), computed redundantly in both halves ---
            float g[12];
            const float* kr = knots + i * NKNOT;
            #pragma unroll
            for (int t = 0; t < 6; ++t) g[3 + t] = kr[t];
            const float hh = (g[8] - g[3]) * 0.2f;
            g[2] = g[3] - hh;  g[1] = g[3] - 2.f * hh;  g[0] = g[3] - 3.f * hh;
            g[9] = g[8] + hh;  g[10] = g[8] + 2.f * hh; g[11] = g[8] + 3.f * hh;

            const float xv = x[(b_base + m) * NUM_IN + i];

            float Bv[11];
            #pragma unroll
            for (int mm = 0; mm < 11; ++mm)
                Bv[mm] = (xv >= g[mm] && xv < g[mm + 1]) ? 1.0f : 0.0f;
            #pragma unroll
            for (int p = 1; p <= 3; ++p) {
                // reciprocals of all distinct denominators for this degree pass:
                // rden[mm] = 1/(g[mm+p]-g[mm]); t2's denom at mm == t1's denom at mm+1
                float rden[11];
                #pragma unroll
                for (int mm = 0; mm + p < 12; ++mm)
                    rden[mm] = __builtin_amdgcn_rcpf(g[mm + p] - g[mm]);
                #pragma unroll
                for (int mm = 0; mm + p < 11; ++mm) {
                    const float t1 = (xv - g[mm]) * rden[mm];
                    const float t2 = (g[mm + p + 1] - xv) * rden[mm + 1];
                    Bv[mm] = t1 * Bv[mm] + t2 * Bv[mm + 1];
                }
            }
            const float sv = xv * __builtin_amdgcn_rcpf(1.0f + __expf(-xv));   // silu(x)

            // --- A fragments: coef row for s=(o_base+m, i), scaled ---
            const int s = (o_base + m) * NUM_IN + i;
            const float mk = mask[s];
            const float ss = scale_spline[s] * mk;
            const float sb = scale_b[s];
            const float* cp = coef + s * NJ + half * 2;   // K0 = 2*half per f32 A layout
            const float a0 = cp[0], a1 = cp[1];           // j = K0, K0+1
            const float a4 = cp[4], a5 = cp[5];           // j = 4+K0, 4+K0+1

            v2f A1q0 = { a0 * mk, a1 * mk };              // post_splines A = mask*coef
            v2f A1q1 = { a4 * mk, a5 * mk };
            v2f A2q0 = { a0 * ss, a1 * ss };              // post_acts A = ss*mask*coef
            v2f A2q1 = { a4 * ss, a5 * ss };
            v2f A2q2 = { half ? 0.f : sb, 0.f };          // j=8 column = scale_b, j=9..11 = 0

            // --- B fragments: basis rows, N = b = m ---
            v2f Bq0 = { half ? Bv[2] : Bv[0], half ? Bv[3] : Bv[1] };
            v2f Bq1 = { half ? Bv[6] : Bv[4], half ? Bv[7] : Bv[5] };
            v2f Bq2 = { half ? 0.f   : sv,    0.f };      // j=8 row = silu(x)

            // --- GEMMs: D1 = post_splines tile, D2 = post_acts tile ---
            v8f D1 = {};
            D1 = __builtin_amdgcn_wmma_f32_16x16x4_f32(false, A1q0, false, Bq0, (short)0, D1, false, false);
            D1 = __builtin_amdgcn_wmma_f32_16x16x4_f32(false, A1q1, false, Bq1, (short)0, D1, false, false);
            v8f D2 = {};
            D2 = __builtin_amdgcn_wmma_f32_16x16x4_f32(false, A2q0, false, Bq0, (short)0, D2, false, false);
            D2 = __builtin_amdgcn_wmma_f32_16x16x4_f32(false, A2q1, false, Bq1, (short)0, D2, false, false);
            D2 = __builtin_amdgcn_wmma_f32_16x16x4_f32(false, A2q2, false, Bq2, (short)0, D2, false, false);

            // --- y partial: sum post_acts over the 16 o rows of this tile ---
            float yr = D2[0] + D2[1] + D2[2] + D2[3] + D2[4] + D2[5] + D2[6] + D2[7];
            yr += __shfl_xor(yr, 16);
            if (lane < 16)
                ws_part[o_tile * 65536 + (b_base + lane) * NUM_IN + i] = yr;

            // --- stage D1 into LDS, hold D2 in registers ---
            #pragma unroll
            for (int r = 0; r < 8; ++r) {
                const int oo = r + half * 8;
                buf[il * IL_STRIDE + oo * OO_STRIDE + m] = D1[r];
            }
            d2h[c] = D2;
        }
        __syncthreads();

        // ---- coalesced store pass 1: post_splines + pre_acts (128B runs over i) ----
        for (int it = 0; it < 32; ++it) {
            const int p  = it * 8 + wave;
            const int bb = p >> 4, oo = p & 15;
            const int gi = (b_base + bb) * SIZE_S + (o_base + oo) * NUM_IN + k * 32 + lane;
            out_spl[gi] = buf[lane * IL_STRIDE + oo * OO_STRIDE + bb];
            out_pre[gi] = x[(b_base + bb) * NUM_IN + k * 32 + lane];
        }
        __syncthreads();

        // ---- stage D2, then store pass 2: post_acts ----
        #pragma unroll
        for (int c = 0; c < 4; ++c) {
            const int il = wave * 4 + c;
            #pragma unroll
            for (int r = 0; r < 8; ++r) {
                const int oo = r + half * 8;
                buf[il * IL_STRIDE + oo * OO_STRIDE + m] = d2h[c][r];
            }
        }
        __syncthreads();
        for (int it = 0; it < 32; ++it) {
            const int p  = it * 8 + wave;
            const int bb = p >> 4, oo = p & 15;
            const int gi = (b_base + bb) * SIZE_S + (o_base + oo) * NUM_IN + k * 32 + lane;
            out_act[gi] = buf[lane * IL_STRIDE + oo * OO_STRIDE + bb];
        }
        __syncthreads();
    }
}

// y[b*128+i] = sum over the 8 o-tile partials
__global__ __launch_bounds__(256)
void kan_reduce(const float* __restrict__ ws_part, float* __restrict__ y)
{
    const int t = blockIdx.x * 256 + threadIdx.x;
    float s = 0.f;
    #pragma unroll
    for (int ot = 0; ot < 8; ++ot) s += ws_part[ot * 65536 + t];
    y[t] = s;
}

extern "C" void kernel_launch(void* const* d_in, const int* in_sizes, int n_in,
                              void* d_out, int out_size, void* d_ws, size_t ws_size,
                              hipStream_t stream) {
    const float* x            = (const float*)d_in[0];
    const float* scale_b      = (const float*)d_in[1];
    const float* scale_spline = (const float*)d_in[2];
    const float* coef         = (const float*)d_in[3];
    const float* mask         = (const float*)d_in[4];
    const float* knots        = (const float*)d_in[5];

    float* out  = (float*)d_out;
    float* y    = out;                       // 512*128
    float* pre  = y + 65536;                 // 512*16384
    float* spl  = pre + PRE_ELEMS;           // 512*16384
    float* act  = spl + PRE_ELEMS;           // 512*16384
    float* part = (float*)d_ws;              // 8 * 512 * 128 floats = 2 MB

    kan_main<<<dim3(512), dim3(256), 0, stream>>>(x, scale_b, scale_spline, coef,
                                                  mask, knots, pre, spl, act, part);
    kan_reduce<<<dim3(256), dim3(256), 0, stream>>>(part, y);
}